// CatmullRomSpline2D_16947940950337
// MI455X (gfx1250) — compile-verified
//
#include <hip/hip_runtime.h>

typedef float v2f __attribute__((ext_vector_type(2)));
typedef float v8f __attribute__((ext_vector_type(8)));

#define GRID_HW 512
#define D_OUT   16
#define PTS_PER_BLOCK 256

// Catmull-Rom basis: (H@C)[::-1]  -> row k = coefficients of s^k
__constant__ float c_basis[4][4] = {
    { 0.0f,  1.0f,  0.0f,  0.0f},
    {-0.5f,  0.0f,  0.5f,  0.0f},
    { 1.0f, -2.5f,  2.0f, -0.5f},
    {-0.5f,  1.5f, -1.5f,  0.5f},
};

__global__ __launch_bounds__(PTS_PER_BLOCK)
void spline2d_kernel(const float2* __restrict__ pts,
                     const float*  __restrict__ cp,
                     float*        __restrict__ out,
                     int npts)
{
    __shared__ float s_cx [PTS_PER_BLOCK][4];
    __shared__ float s_cy [PTS_PER_BLOCK][4];
    __shared__ int   s_yof[PTS_PER_BLOCK][4];
    __shared__ int   s_xof[PTS_PER_BLOCK][4];

    const int t         = threadIdx.x;
    const int blockBase = blockIdx.x * PTS_PER_BLOCK;
    const int p         = blockBase + t;

    // branch-free clamped load (keeps EXEC all-ones for the WMMAs below)
    const int psafe = (p < npts) ? p : (npts - 1);
    const float2 xy = pts[psafe];
    const float x = xy.x, y = xy.y;

    const int ix0 = (int)x;
    const int iy0 = (int)y;
    const float sx = x - floorf(x);
    const float sy = y - floorf(y);

    // ---- stage 1a: clamped cell offsets (element units) ----
#pragma unroll
    for (int r = 0; r < 4; ++r) {
        int iy = iy0 - 1 + r; iy = iy < 0 ? 0 : (iy > GRID_HW - 1 ? GRID_HW - 1 : iy);
        int ix = ix0 - 1 + r; ix = ix < 0 ? 0 : (ix > GRID_HW - 1 ? GRID_HW - 1 : ix);
        s_yof[t][r] = iy * (GRID_HW * D_OUT);
        s_xof[t][r] = ix * D_OUT;
    }

    // ---- stage 1b: cubic weights via V_WMMA_F32_16X16X4_F32 ----
    // D(16x16) = A(16 pts x 4 powers) x B(4 x 16 basis, cols 4..15 = 0)
    const int  lane     = t & 31;
    const int  waveBase = t & ~31;   // wave's first local point
    const int  n        = lane & 15; // matrix column held by this lane
    const bool hiHalf   = lane >= 16;

    // B layout: row K striped across lanes in one VGPR;
    // VGPR0 = K0 (lanes 0-15) / K2 (lanes 16-31), VGPR1 = K1 / K3
    v2f Bm;
    Bm.x = (n < 4) ? c_basis[hiHalf ? 2 : 0][n] : 0.0f;
    Bm.y = (n < 4) ? c_basis[hiHalf ? 3 : 1][n] : 0.0f;

#pragma unroll
    for (int h = 0; h < 2; ++h) {           // two 16-point halves of the wave
        const int src = h * 16 + n;         // lane owning point (row) n of this half
        {   // x weights
            float s  = __shfl(sx, src, 32);
            float s2 = s * s, s3 = s2 * s;
            // A layout: lane m holds row m: VGPR0 = {s^0 | s^2}, VGPR1 = {s^1 | s^3}
            v2f Am; Am.x = hiHalf ? s2 : 1.0f; Am.y = hiHalf ? s3 : s;
            v8f Cm = {};
            v8f Dm = __builtin_amdgcn_wmma_f32_16x16x4_f32(
                         false, Am, false, Bm, (short)0, Cm, false, false);
            if (n < 4) {
#pragma unroll
                for (int c = 0; c < 8; ++c) {
                    int m = c + (hiHalf ? 8 : 0);      // D row index
                    s_cx[waveBase + h * 16 + m][n] = Dm[c];
                }
            }
        }
        {   // y weights
            float s  = __shfl(sy, src, 32);
            float s2 = s * s, s3 = s2 * s;
            v2f Am; Am.x = hiHalf ? s2 : 1.0f; Am.y = hiHalf ? s3 : s;
            v8f Cm = {};
            v8f Dm = __builtin_amdgcn_wmma_f32_16x16x4_f32(
                         false, Am, false, Bm, (short)0, Cm, false, false);
            if (n < 4) {
#pragma unroll
                for (int c = 0; c < 8; ++c) {
                    int m = c + (hiHalf ? 8 : 0);
                    s_cy[waveBase + h * 16 + m][n] = Dm[c];
                }
            }
        }
    }

    __syncthreads();

    // ---- stage 2: coalesced vector gather+accumulate ----
    // 4 lanes per point, each lane owns a channel quad -> one b128 per tap/lane,
    // 4 lanes together fetch the full 64B cell line.
    const int d4 = (t & 3) * 4;  // first channel of this lane's quad
    const int g  = t >> 2;       // point group 0..63

#pragma unroll 1
    for (int tile = 0; tile < 4; ++tile) {
        const int lp = tile * 64 + g;            // local point 0..255
        const int pp = blockBase + lp;
        float a0 = 0.0f, a1 = 0.0f, a2 = 0.0f, a3 = 0.0f;
#pragma unroll
        for (int i = 0; i < 4; ++i) {
            const float wy = s_cy[lp][i];
            const int   ro = s_yof[lp][i];
#pragma unroll
            for (int j = 0; j < 4; ++j) {
                const float  w   = wy * s_cx[lp][j];
                const int    off = ro + s_xof[lp][j] + d4;
                const float4 v   = *(const float4*)(cp + off);
                a0 = fmaf(w, v.x, a0);
                a1 = fmaf(w, v.y, a1);
                a2 = fmaf(w, v.z, a2);
                a3 = fmaf(w, v.w, a3);
            }
        }
        if (pp < npts) {
            *(float4*)(out + pp * D_OUT + d4) = make_float4(a0, a1, a2, a3);
        }
    }
}

extern "C" void kernel_launch(void* const* d_in, const int* in_sizes, int n_in,
                              void* d_out, int out_size, void* d_ws, size_t ws_size,
                              hipStream_t stream) {
    (void)n_in; (void)out_size; (void)d_ws; (void)ws_size;
    const float2* pts = (const float2*)d_in[0];
    const float*  cp  = (const float*)d_in[1];
    float*        out = (float*)d_out;
    const int npts    = in_sizes[0] / 2;
    const int nblocks = (npts + PTS_PER_BLOCK - 1) / PTS_PER_BLOCK;
    spline2d_kernel<<<nblocks, PTS_PER_BLOCK, 0, stream>>>(pts, cp, out, npts);
}